// FFF_120259084544
// MI455X (gfx1250) — compile-verified
//
#include <hip/hip_runtime.h>

#define N_IN   1024
#define N_OUT  1024
#define DEPTH  10
#define SPB    16      // samples per block
#define THREADS 128
#define ROW_PITCH 1040 // 1024 + 4 pad DW per 256 DW (TDM pad_interval=256DW, pad_amount=4DW)

typedef float v2f __attribute__((ext_vector_type(2)));
typedef float v8f __attribute__((ext_vector_type(8)));
typedef unsigned int u32;
typedef unsigned long long u64;
typedef u32 u32x4 __attribute__((ext_vector_type(4)));
typedef u32 u32x8 __attribute__((ext_vector_type(8)));

__global__ __launch_bounds__(THREADS) void fff_kernel(
    const float* __restrict__ x,
    const float* __restrict__ Xw,
    const float* __restrict__ Yw,
    float* __restrict__ out)
{
  extern __shared__ __align__(16) float smem[];
  float* xs     = smem;                       // SPB * ROW_PITCH floats
  float* lamAll = smem + SPB * ROW_PITCH;     // DEPTH * SPB
  int*   nodeAll = (int*)(lamAll + DEPTH * SPB); // DEPTH * SPB

  const int tid  = threadIdx.x;
  const int wave = tid >> 5;
  const int lane = tid & 31;
  const u64 rowBase = (u64)blockIdx.x * SPB;

  // ---------- Phase 0: TDM bulk copy of the x tile into LDS ----------
  if (wave == 0) {
    u32 ldsAddr = (u32)(size_t)(void*)xs;     // low 32 bits of generic LDS addr = LDS offset
    u64 ga = (u64)(size_t)(const void*)(x + rowBase * N_IN);
    u32x4 g0;
    g0[0] = 1u;                               // count=1, is_restore=0, gather_mode=0
    g0[1] = ldsAddr;                          // lds_addr
    g0[2] = (u32)ga;                          // global_addr[31:0]
    g0[3] = ((u32)(ga >> 32) & 0x01FFFFFFu) | 0x80000000u; // global_addr[56:32], type=2
    u32x8 g1;
    g1[0] = (2u << 16)      // data_size = 4B
          | (1u << 20)      // pad_enable
          | (7u << 22)      // pad_interval = 256 DWORDs
          | (3u << 25);     // pad_amount   = 4 DWORDs
    g1[1] = (u32)(N_IN) << 16;        // atomic_barrier_addr=0 | tensor_dim0[15:0]=1024
    g1[2] = ((u32)SPB) << 16;         // tensor_dim0[31:16]=0 | tensor_dim1[15:0]=16
    g1[3] = ((u32)N_IN) << 16;        // tensor_dim1[31:16]=0 | tile_dim0=1024
    g1[4] = (u32)SPB;                 // tile_dim1=16, tile_dim2=0
    g1[5] = (u32)N_IN;                // tensor_dim0_stride[31:0]=1024
    g1[6] = 0u;                       // stride hi, tensor_dim1_stride lo
    g1[7] = 0u;
    asm volatile("tensor_load_to_lds %0, %1" :: "s"(g0), "s"(g1) : "memory");
    __builtin_amdgcn_s_wait_tensorcnt(0);
  }
  __syncthreads();

  // ---------- Phase 1: tree routing (per-sample dots, 8 lanes/sample) ----------
  {
    const int s = (wave << 2) + (lane >> 3);  // sample 0..15
    const int j = lane & 7;                   // lane-in-group
    int node = 0;
    for (int d = 0; d < DEPTH; ++d) {
      const float4* Xr = (const float4*)(Xw + (size_t)node * N_IN);
      float acc = 0.0f;
      #pragma unroll 8
      for (int k4 = 0; k4 < 32; ++k4) {
        int e4 = j + (k4 << 3);               // float4 index 0..255
        int e  = e4 << 2;                     // element index
        int lo = e + ((e >> 8) << 2);         // TDM-padded LDS offset
        float4 xv = *(const float4*)(xs + s * ROW_PITCH + lo);
        float4 wv = Xr[e4];
        acc += xv.x * wv.x + xv.y * wv.y + xv.z * wv.z + xv.w * wv.w;
      }
      acc += __shfl_xor(acc, 1, 32);
      acc += __shfl_xor(acc, 2, 32);
      acc += __shfl_xor(acc, 4, 32);          // all 8 lanes hold lam
      if (j == 0) { lamAll[d * SPB + s] = acc; nodeAll[d * SPB + s] = node; }
      node = (node << 1) + 1 + (acc > 0.0f ? 1 : 0);
    }
  }
  __syncthreads();

  // ---------- Phase 2: y_tile += diag(lam_d) x Ygathered_d via WMMA f32 16x16x4 ----------
  {
    const int m     = lane & 15;              // N index (column within chunk) and M index
    const int khalf = lane >> 4;              // 0: K={0,1}, 1: K={2,3}
    for (int t = 0; t < 16; ++t) {
      const int colbase = (wave * 16 + t) * 16;   // this wave's 16-column chunk
      v8f acc = {};
      for (int d = 0; d < DEPTH; ++d) {
        float lamv = lamAll[d * SPB + m];
        #pragma unroll
        for (int kk = 0; kk < 4; ++kk) {
          int r0 = 4 * kk + (khalf << 1);     // diag rows covered by this K-chunk
          int r1 = r0 + 1;
          v2f a;
          a.x = (m == r0) ? lamv : 0.0f;      // A[m][2*khalf+0]
          a.y = (m == r1) ? lamv : 0.0f;      // A[m][2*khalf+1]
          int n0 = nodeAll[d * SPB + r0];
          int n1 = nodeAll[d * SPB + r1];
          v2f b;
          b.x = Yw[(size_t)n0 * N_OUT + colbase + m];   // B[2*khalf+0][m]
          b.y = Yw[(size_t)n1 * N_OUT + colbase + m];   // B[2*khalf+1][m]
          acc = __builtin_amdgcn_wmma_f32_16x16x4_f32(
              false, a, false, b, (short)0, acc, false, false);
        }
      }
      // D layout: lane holds N=m, M = 8*khalf + v
      #pragma unroll
      for (int v = 0; v < 8; ++v) {
        out[(rowBase + (khalf << 3) + v) * N_OUT + colbase + m] = acc[v];
      }
    }
  }
}

extern "C" void kernel_launch(void* const* d_in, const int* in_sizes, int n_in,
                              void* d_out, int out_size, void* d_ws, size_t ws_size,
                              hipStream_t stream) {
  const float* x  = (const float*)d_in[0];
  const float* Xw = (const float*)d_in[1];
  const float* Yw = (const float*)d_in[2];
  float* out = (float*)d_out;
  int n_batch = in_sizes[0] / N_IN;
  int blocks  = n_batch / SPB;
  size_t smem = (size_t)(SPB * ROW_PITCH + DEPTH * SPB) * sizeof(float)
              + (size_t)(DEPTH * SPB) * sizeof(int);
  fff_kernel<<<blocks, THREADS, smem, stream>>>(x, Xw, Yw, out);
}